// BaseInformationLoss_5918464934403
// MI455X (gfx1250) — compile-verified
//
#include <hip/hip_runtime.h>
#include <hip/hip_bf16.h>

typedef __attribute__((ext_vector_type(2))) float v2f;
typedef __attribute__((ext_vector_type(8))) float v8f;

#define B_PTS   1024
#define N_DIM   64
#define EPS_RHO 1e-12f
#define EPS_LOG 1e-8f

// ---- workspace layout (floats) ----
#define WS_H      0      // h[64]        per-dim KDE bandwidth
#define WS_NORM   64     // norm[1024]   per-row squared norm
#define WS_MARG   1088   // marg_acc[64] sum over q of log(density)
#define WS_KNN    1152   // knn_acc      sum over i of log(rho_sq)

// Zero the accumulators (poisoned workspace, replayed graph => must re-init).
__global__ void il_init_kernel(float* acc) {
    int t = blockIdx.x * blockDim.x + threadIdx.x;
    if (t < 65) acc[t] = 0.0f;   // acc = ws + WS_MARG  (64 marg + 1 knn)
}

// Per-dim mean/std -> bandwidth h[r].  One block per dim.
__global__ __launch_bounds__(256) void il_stats_kernel(const float* __restrict__ act,
                                                       float* __restrict__ hArr) {
    __shared__ float s1[256], s2[256];
    const int r = blockIdx.x, tid = threadIdx.x;
    float a = 0.0f, b = 0.0f;
    for (int j = tid; j < B_PTS; j += 256) {
        float v = act[j * N_DIM + r];
        a += v; b += v * v;
    }
    s1[tid] = a; s2[tid] = b;
    __syncthreads();
    for (int off = 128; off > 0; off >>= 1) {
        if (tid < off) { s1[tid] += s1[tid + off]; s2[tid] += s2[tid + off]; }
        __syncthreads();
    }
    if (tid == 0) {
        float sum = s1[0], sq = s2[0];
        float var = (sq - sum * sum * (1.0f / B_PTS)) * (1.0f / (B_PTS - 1));
        var = fmaxf(var, 0.0f);
        // 1024^(-0.2) == 2^(-2) == 0.25 exactly
        float h = fmaxf(1.06f * sqrtf(var) * 0.25f, 1e-4f);
        hArr[r] = h;
    }
}

// Per-row squared norms.
__global__ __launch_bounds__(128) void il_norms_kernel(const float* __restrict__ act,
                                                       float* __restrict__ norms) {
    int i = blockIdx.x * blockDim.x + threadIdx.x;
    if (i < B_PTS) {
        const float4* row = (const float4*)(act + i * N_DIM);
        float s = 0.0f;
        #pragma unroll
        for (int n = 0; n < N_DIM / 4; ++n) {
            float4 v = row[n];
            s += v.x * v.x + v.y * v.y + v.z * v.z + v.w * v.w;
        }
        norms[i] = s;
    }
}

// Reflected-Gaussian KDE: block = (q-tile of 256, fixed dim r).
// Sample column staged in LDS; inner loop is 3x v_exp_f32 per sample,
// unrolled x4 via float4 LDS reads for TRANS/VALU overlap.
__device__ __forceinline__ float kde_term(float x, float y, float inv_h) {
    float u1 = (x - y) * inv_h;
    float u2 = (x + y) * inv_h;
    float u3 = (x - 2.0f + y) * inv_h;
    return __expf(-0.5f * u1 * u1) + __expf(-0.5f * u2 * u2) + __expf(-0.5f * u3 * u3);
}

__global__ __launch_bounds__(256) void il_kde_kernel(const float* __restrict__ act,
                                                     const float* __restrict__ hArr,
                                                     float* __restrict__ marg_acc) {
    __shared__ float sY[B_PTS];
    __shared__ float red[256];
    const int r = blockIdx.y;
    const int tid = threadIdx.x;
    for (int j = tid; j < B_PTS; j += 256) sY[j] = act[j * N_DIM + r];
    __syncthreads();

    const float h = hArr[r];
    const float inv_h = 1.0f / h;
    const int q = blockIdx.x * 256 + tid;
    const float x = sY[q];                       // query == sample, already staged

    const float4* sY4 = (const float4*)sY;
    float s = 0.0f;
    #pragma unroll 2
    for (int j4 = 0; j4 < B_PTS / 4; ++j4) {
        float4 y = sY4[j4];
        s += kde_term(x, y.x, inv_h);
        s += kde_term(x, y.y, inv_h);
        s += kde_term(x, y.z, inv_h);
        s += kde_term(x, y.w, inv_h);
    }
    const float inv_sqrt2pi = 0.3989422804014327f;
    float density = s * inv_sqrt2pi * inv_h * (1.0f / B_PTS);
    red[tid] = logf(density + EPS_LOG);
    __syncthreads();
    for (int off = 128; off > 0; off >>= 1) {
        if (tid < off) red[tid] += red[tid + off];
        __syncthreads();
    }
    if (tid == 0) atomicAdd(&marg_acc[r], red[0]);
}

// B-panel fragment loader for the 16x16x4 f32 WMMA chain.
// Lane holds row (j16+lo); K-step s needs float2 index (2s + hi).
__device__ __forceinline__ void load_bfrag(v2f* __restrict__ dst,
                                           const float* __restrict__ act,
                                           int j16, int hi, int lo) {
    const float2* brow = (const float2*)(act + (size_t)(j16 + lo) * N_DIM);
    #pragma unroll
    for (int s = 0; s < 16; ++s) {
        float2 t = brow[2 * s + hi];
        dst[s].x = t.x; dst[s].y = t.y;
    }
}

// kNN via WMMA Gram matrix, fused with per-row top-6 selection.
// One wave per 16-row strip; B panel is register double-pumped: next
// tile's 16 loads are issued right after the WMMA chain and their latency
// is covered by the distance epilogue + LDS selection (two barriers).
__global__ __launch_bounds__(32) void il_knn_kernel(const float* __restrict__ act,
                                                    const float* __restrict__ norms,
                                                    const int* __restrict__ kptr,
                                                    float* __restrict__ knn_acc) {
    __shared__ float tile[16][17];
    __shared__ float normI[16];

    const int lane = threadIdx.x;     // 0..31
    const int hi   = lane >> 4;       // 0/1 -> K half
    const int lo   = lane & 15;
    const int i16  = blockIdx.x * 16;

    // Preload A fragments: lane holds row (i16+lo); K-step s needs
    // floats [4s + 2*hi, 4s + 2*hi + 1] == float2 index (2s + hi).
    v2f afrag[16];
    load_bfrag(afrag, act, i16, hi, lo);
    if (lane < 16) normI[lane] = norms[i16 + lane];
    __syncthreads();

    float best[6];
    #pragma unroll
    for (int t = 0; t < 6; ++t) best[t] = 3.4e38f;

    v2f bfrag[16];
    load_bfrag(bfrag, act, 0, hi, lo);            // prologue: tile jt=0

    for (int jt = 0; jt < B_PTS / 16; ++jt) {
        const int j16 = jt * 16;

        v8f c = {};
        #pragma unroll
        for (int s = 0; s < 16; ++s) {
            c = __builtin_amdgcn_wmma_f32_16x16x4_f32(
                    /*neg_a=*/false, afrag[s],
                    /*neg_b=*/false, bfrag[s],
                    /*c_mod=*/(short)0, c,
                    /*reuse_a=*/false, /*reuse_b=*/false);
        }

        // Prefetch next tile's B panel; latency hidden by the epilogue below.
        if (jt + 1 < B_PTS / 16)
            load_bfrag(bfrag, act, j16 + 16, hi, lo);

        // Epilogue: convert Gram to distances into LDS tile.
        float nJ = norms[j16 + lo];               // this lane's column n = lo
        #pragma unroll
        for (int v = 0; v < 8; ++v) {
            int m = v + 8 * hi;
            tile[m][lo] = normI[m] + nJ - 2.0f * c[v];
        }
        __syncthreads();

        // Lanes 0..15 each own row m=lane: merge 16 candidates into top-6.
        if (lane < 16) {
            #pragma unroll
            for (int n = 0; n < 16; ++n) {
                float d = tile[lane][n];
                if (d < best[5]) {
                    best[5] = d;
                    #pragma unroll
                    for (int t = 5; t > 0; --t) {
                        if (best[t] < best[t - 1]) {
                            float tmp = best[t]; best[t] = best[t - 1]; best[t - 1] = tmp;
                        }
                    }
                }
            }
        }
        __syncthreads();
    }

    if (lane < 16) {
        int kk = *kptr;               // reference: rho_sq = (k+1)-th smallest (0-based idx k)
        if (kk > 5) kk = 5;
        float rho = best[5];
        #pragma unroll
        for (int t = 0; t < 6; ++t) if (kk == t) rho = best[t];
        rho = fmaxf(rho, EPS_RHO);
        atomicAdd(knn_acc, logf(rho));
    }
}

// Finalize: entropy constants in double, write 65 outputs.
__global__ void il_final_kernel(const float* __restrict__ marg_acc,
                                const float* __restrict__ knn_acc,
                                const int* __restrict__ kptr,
                                float* __restrict__ out) {
    int t = blockIdx.x * blockDim.x + threadIdx.x;
    if (t < 64) out[1 + t] = -marg_acc[t] * (1.0f / B_PTS);
    if (t == 64) {
        int k = *kptr;
        double dg_k = -0.57721566490153286;        // digamma(k) = -gamma + H_{k-1}
        for (int m = 1; m < k; ++m) dg_k += 1.0 / (double)m;
        const double Bd = (double)B_PTS, Nd = (double)N_DIM;
        double dg_B = log(Bd) - 1.0 / (2.0 * Bd) - 1.0 / (12.0 * Bd * Bd);
        double log_c_d = 0.5 * Nd * log(3.14159265358979323846) - lgamma(0.5 * Nd + 1.0);
        double mean_log = (double)(*knn_acc) / Bd;
        double h_nats = -dg_k + dg_B + log_c_d + Nd * (0.5 * mean_log);
        out[0] = (float)(h_nats / 0.6931471805599453);
    }
}

extern "C" void kernel_launch(void* const* d_in, const int* in_sizes, int n_in,
                              void* d_out, int out_size, void* d_ws, size_t ws_size,
                              hipStream_t stream) {
    const float* act  = (const float*)d_in[0];
    const int*   kptr = (const int*)d_in[1];
    float* out = (float*)d_out;
    float* ws  = (float*)d_ws;

    float* h        = ws + WS_H;
    float* norm     = ws + WS_NORM;
    float* marg_acc = ws + WS_MARG;
    float* knn_acc  = ws + WS_KNN;

    il_init_kernel<<<1, 128, 0, stream>>>(marg_acc);              // zeros marg+knn (65 floats)
    il_stats_kernel<<<N_DIM, 256, 0, stream>>>(act, h);
    il_norms_kernel<<<(B_PTS + 127) / 128, 128, 0, stream>>>(act, norm);
    il_kde_kernel<<<dim3(B_PTS / 256, N_DIM), 256, 0, stream>>>(act, h, marg_acc);
    il_knn_kernel<<<B_PTS / 16, 32, 0, stream>>>(act, norm, kptr, knn_acc);
    il_final_kernel<<<1, 128, 0, stream>>>(marg_acc, knn_acc, kptr, out);
}